// SpectrogramExtractor_29669634081057
// MI455X (gfx1250) — compile-verified
//
#include <hip/hip_runtime.h>
#include <hip/hip_bf16.h>
#include <math.h>

typedef __attribute__((ext_vector_type(2))) float v2f;
typedef __attribute__((ext_vector_type(8))) float v8f;

#define HOP 441
#define NMEL 80

// One workgroup: 16 consecutive frames of one batch row, one FFT band.
// Phase 1 (8 waves): frame extract + window + circular shift + radix-2 FFT in
//                    LDS + power spectrum rows (zero-padded K to multiple of 4).
// Phase 2 (waves 0..4): mel projection as WMMA f32 16x16x4 GEMM, log, store.
template <int L, int LOG2L>
__global__ __launch_bounds__(256) void band_kernel(
    const float* __restrict__ x, const float* __restrict__ window,
    const float* __restrict__ mel, float* __restrict__ out,
    int N, int T, int band) {
  constexpr int HALF = L / 2;
  constexpr int ROWS = HALF + 4;  // K padded to multiple of 4

  extern __shared__ float smem[];
  float* re = smem;            // [L]
  float* im = smem + L;        // [L]
  float* pw = smem + 2 * L;    // [16][ROWS]

  const int tid = threadIdx.x;
  const int nthr = blockDim.x;
  const int t0 = blockIdx.x * 16;
  const int b = blockIdx.y;
  const float* xb = x + (long long)b * N;
  const long long padded_len = (long long)N + HALF;

  // ---------------- Phase 1: 16 FFTs ----------------
  for (int f16 = 0; f16 < 16; ++f16) {
    const long long start = (long long)(t0 + f16) * HOP;
    long long pad_amt = (long long)L - (padded_len - start);
    if (pad_amt < 0) pad_amt = 0;

    // load (bit-reversed) with window + circular shift by HALF
    for (int j = tid; j < L; j += nthr) {
      const int jj = (j + HALF) & (L - 1);
      const long long idx = start + (long long)jj - pad_amt;  // index into front-padded wav
      const long long xi = idx - HALF;
      float v = 0.0f;
      if (xi >= 0 && xi < (long long)N) v = xb[xi] * window[jj];
      const int r = (int)(__brev((unsigned)j) >> (32 - LOG2L));
      re[r] = v;
      im[r] = 0.0f;
    }
    __syncthreads();

    // iterative in-place radix-2 DIT FFT (forward)
    for (int len = 2; len <= L; len <<= 1) {
      const int hl = len >> 1;
      const float ang = -6.283185307179586f / (float)len;
      for (int k = tid; k < (L >> 1); k += nthr) {
        const int m = k & (hl - 1);
        const int i = ((k & ~(hl - 1)) << 1) | m;
        const int j = i + hl;
        float s, c;
        __sincosf(ang * (float)m, &s, &c);
        const float xr = re[j], xii = im[j];
        const float tr = c * xr - s * xii;
        const float ti = c * xii + s * xr;
        const float ur = re[i], ui = im[i];
        re[j] = ur - tr; im[j] = ui - ti;
        re[i] = ur + tr; im[i] = ui + ti;
      }
      __syncthreads();
    }

    // power spectrum row, zero-padded to ROWS
    float* prow = pw + f16 * ROWS;
    for (int f = tid; f < ROWS; f += nthr) {
      float p = 0.0f;
      if (f <= HALF) {
        const float rr = re[f], ii2 = im[f];
        p = rr * rr + ii2 * ii2;
      }
      prow[f] = p;
    }
    __syncthreads();
  }

  // ---------------- Phase 2: mel GEMM via WMMA f32 16x16x4 ----------------
  const int w = tid >> 5;      // wave id (wave32)
  const int lane = tid & 31;
  if (w < 5) {                 // 5 tiles of 16 mel columns = 80 (wave-uniform)
    const int m0 = w * 16;
    const int col = m0 + (lane & 15);
    const int koff = (lane < 16) ? 0 : 2;             // A/B second-half-lanes K offset
    const float* prow = pw + (lane & 15) * ROWS;      // A row = frame index

    v8f acc = {0.f, 0.f, 0.f, 0.f, 0.f, 0.f, 0.f, 0.f};
    for (int k0 = 0; k0 < ROWS; k0 += 4) {
      v2f a;
      a.x = prow[k0 + koff];
      a.y = prow[k0 + koff + 1];
      int k1 = k0 + koff;
      int k2 = k1 + 1;
      if (k1 > HALF) k1 = HALF;   // padded-K: A is zero there, clamp B index in-bounds
      if (k2 > HALF) k2 = HALF;
      v2f bm;
      bm.x = mel[(long long)k1 * NMEL + col];
      bm.y = mel[(long long)k2 * NMEL + col];
      acc = __builtin_amdgcn_wmma_f32_16x16x4_f32(
          /*neg_a=*/false, a, /*neg_b=*/false, bm,
          /*c_mod=*/(short)0, acc, /*reuse_a=*/false, /*reuse_b=*/false);
    }

    // D layout: VGPR r -> M = r (+8 for lanes 16..31), N = lane & 15
    const int mrow_base = (lane < 16) ? 0 : 8;
#pragma unroll
    for (int r = 0; r < 8; ++r) {
      const int t = t0 + mrow_base + r;
      if (t < T) {
        const size_t o = (((size_t)b * T + t) * NMEL + col) * 3 + band;
        out[o] = logf(acc[r] + 1e-16f);
      }
    }
  }
}

extern "C" void kernel_launch(void* const* d_in, const int* in_sizes, int n_in,
                              void* d_out, int out_size, void* d_ws, size_t ws_size,
                              hipStream_t stream) {
  const float* x  = (const float*)d_in[0];
  const float* w1 = (const float*)d_in[1];
  const float* w2 = (const float*)d_in[2];
  const float* w4 = (const float*)d_in[3];
  const float* m1 = (const float*)d_in[4];
  const float* m2 = (const float*)d_in[5];
  const float* m4 = (const float*)d_in[6];
  float* out = (float*)d_out;

  // Infer (B, N) from flat sizes: out_size == B * T * 80 * 3, T = ceil((N+512)/441)
  const long long S0 = in_sizes[0];
  int B = 4;
  long long N = S0 / 4;
  for (int bb = 1; bb <= 64; ++bb) {
    if (S0 % bb) continue;
    const long long n = S0 / bb;
    const long long t = (n + 512 + HOP - 1) / HOP;
    if ((long long)bb * t * NMEL * 3 == (long long)out_size) { B = bb; N = n; break; }
  }
  const int T = (int)((N + 512 + HOP - 1) / HOP);
  const int tilesT = (T + 15) / 16;

  const size_t lds1 = (size_t)(2 * 1024 + 16 * (512 + 4)) * 4;
  const size_t lds2 = (size_t)(2 * 2048 + 16 * (1024 + 4)) * 4;
  const size_t lds4 = (size_t)(2 * 4096 + 16 * (2048 + 4)) * 4;

  (void)hipFuncSetAttribute(reinterpret_cast<const void*>(&band_kernel<1024, 10>),
                            hipFuncAttributeMaxDynamicSharedMemorySize, (int)lds1);
  (void)hipFuncSetAttribute(reinterpret_cast<const void*>(&band_kernel<2048, 11>),
                            hipFuncAttributeMaxDynamicSharedMemorySize, (int)lds2);
  (void)hipFuncSetAttribute(reinterpret_cast<const void*>(&band_kernel<4096, 12>),
                            hipFuncAttributeMaxDynamicSharedMemorySize, (int)lds4);

  dim3 grid(tilesT, B), block(256);
  band_kernel<1024, 10><<<grid, block, lds1, stream>>>(x, w1, m1, out, (int)N, T, 0);
  band_kernel<2048, 11><<<grid, block, lds2, stream>>>(x, w2, m2, out, (int)N, T, 1);
  band_kernel<4096, 12><<<grid, block, lds4, stream>>>(x, w4, m4, out, (int)N, T, 2);
}